// LidarTargets_41850161332646
// MI455X (gfx1250) — compile-verified
//
#include <hip/hip_runtime.h>
#include <hip/hip_bf16.h>
#include <math.h>

// ---- problem constants (from reference) ----
#define BB 16
#define NN 6
#define PP 70000
#define HH 112          // 448/4
#define WW 200          // 800/4
#define GROUPS (PP/16)  // 4375 point-groups of 16 per batch
#define HWSZ (HH*WW)    // 22400

typedef float v2f __attribute__((ext_vector_type(2)));
typedef float v8f __attribute__((ext_vector_type(8)));

// ------------------------------------------------------------------
// Kernel 0: zero the output grid (harness poisons d_out with 0xAA).
// ------------------------------------------------------------------
__global__ void lt_zero_kernel(float* __restrict__ p, int n) {
    int i = blockIdx.x * blockDim.x + threadIdx.x;
    int n4 = n >> 2;
    if (i < n4) {
        float4 z; z.x = 0.f; z.y = 0.f; z.z = 0.f; z.w = 0.f;
        ((float4*)p)[i] = z;
    }
    if (i == 0) {
        for (int k = n4 * 4; k < n; ++k) p[k] = 0.0f;
    }
}

// ------------------------------------------------------------------
// Kernel 1: build 96 fused 4x4 matrices  M = [K4*Einv rows 0..2 ; Einv row2] * Vinv
// Row-major 16 floats per (b,n) into d_ws. M @ [x,y,z,1] = (xpix, ypix, norm, z).
// ------------------------------------------------------------------
__global__ void lt_mat_kernel(const float* __restrict__ E_,
                              const float* __restrict__ K_,
                              const float* __restrict__ V_,
                              float* __restrict__ ws) {
    int bn = blockIdx.x * blockDim.x + threadIdx.x;
    if (bn >= BB * NN) return;
    int b = bn / NN;

    // ---- affine inverse of view ----
    const float* V = V_ + (size_t)b * 16;
    float a00=V[0], a01=V[1], a02=V[2],  t0=V[3];
    float a10=V[4], a11=V[5], a12=V[6],  t1=V[7];
    float a20=V[8], a21=V[9], a22=V[10], t2=V[11];
    float c00 = a11*a22 - a12*a21;
    float c01 = a12*a20 - a10*a22;
    float c02 = a10*a21 - a11*a20;
    float det = a00*c00 + a01*c01 + a02*c02;
    float id  = 1.0f / det;                       // once per (b,n): keep exact
    float i00 = c00*id, i01 = (a02*a21 - a01*a22)*id, i02 = (a01*a12 - a02*a11)*id;
    float i10 = c01*id, i11 = (a00*a22 - a02*a20)*id, i12 = (a02*a10 - a00*a12)*id;
    float i20 = c02*id, i21 = (a01*a20 - a00*a21)*id, i22 = (a00*a11 - a01*a10)*id;
    float Vi[16] = {
        i00, i01, i02, -(i00*t0 + i01*t1 + i02*t2),
        i10, i11, i12, -(i10*t0 + i11*t1 + i12*t2),
        i20, i21, i22, -(i20*t0 + i21*t1 + i22*t2),
        0.f, 0.f, 0.f, 1.f
    };

    // ---- rigid inverse of extrinsics ----
    const float* E = E_ + (size_t)bn * 16;
    float Ei[16];
#pragma unroll
    for (int r = 0; r < 3; ++r) {
#pragma unroll
        for (int c = 0; c < 3; ++c) Ei[r*4 + c] = E[c*4 + r];      // R^T
        Ei[r*4 + 3] = -(E[0*4 + r]*E[3] + E[1*4 + r]*E[7] + E[2*4 + r]*E[11]); // -R^T t
    }
    Ei[12] = 0.f; Ei[13] = 0.f; Ei[14] = 0.f; Ei[15] = 1.f;

    // ---- scaled intrinsics (sx = sy = 1/DS = 0.25) ----
    const float* K = K_ + (size_t)bn * 9;
    float fx = K[0] * 0.25f, fy = K[4] * 0.25f;
    float cx = K[2] * 0.25f, cy = K[5] * 0.25f;

    // Mcam rows: [fx*Ei0 + cx*Ei2 ; fy*Ei1 + cy*Ei2 ; Ei2 ; Ei2]
    float Mc[16];
#pragma unroll
    for (int c = 0; c < 4; ++c) {
        Mc[0*4 + c] = fx * Ei[0*4 + c] + cx * Ei[2*4 + c];
        Mc[1*4 + c] = fy * Ei[1*4 + c] + cy * Ei[2*4 + c];
        Mc[2*4 + c] = Ei[2*4 + c];
        Mc[3*4 + c] = Ei[2*4 + c];
    }

    // Mfull = Mc @ Vi
#pragma unroll
    for (int r = 0; r < 4; ++r) {
#pragma unroll
        for (int c = 0; c < 4; ++c) {
            float s = 0.f;
#pragma unroll
            for (int k = 0; k < 4; ++k) s += Mc[r*4 + k] * Vi[k*4 + c];
            ws[(size_t)bn * 16 + r*4 + c] = s;
        }
    }
}

// ------------------------------------------------------------------
// Epilogue for one (point, camera) result: (xpix, ypix, norm, z)
// Short VALU tail: hardware rcp + hardware log2 (ln(1+v)/ln256 == log2(1+v)/8).
// ------------------------------------------------------------------
__device__ __forceinline__ void lt_emit(float* __restrict__ out, int bn,
                                        float xp, float yp, float nm, float z,
                                        float iluv, float dmax, float inv_bh) {
    float invn = __builtin_amdgcn_rcpf(fmaxf(nm, 1e-6f));   // v_rcp_f32
    float x_ = xp * invn;
    float y_ = yp * invn;
    bool valid = (x_ > -0.5f) && (x_ < (float)WW - 0.5f) &&
                 (y_ > -0.5f) && (y_ < (float)HH - 0.5f) && (z > 0.0f);
    float depth = valid ? fminf(fmaxf(nm, 0.0f), dmax) * inv_bh : 0.0f;
    float vi    = valid ? fminf(fmaxf(iluv, 0.0f), 255.0f) : 0.0f;
    float iout  = __builtin_amdgcn_logf(1.0f + vi) * 0.125f; // log2(1+v)/8
    int ym = (int)fminf(fmaxf(y_, 0.0f), (float)(HH - 1));
    int xm = (int)fminf(fmaxf(x_, 0.0f), (float)(WW - 1));
    size_t base = (size_t)bn * (2 * HWSZ) + (size_t)ym * WW + xm;
    out[base]        = depth;   // depth channel
    out[base + HWSZ] = iout;    // intensity channel
}

// ------------------------------------------------------------------
// Kernel 2: one wave32 per (batch, group-of-16-points).
// Two V_WMMA_F32_16X16X4_F32 per wave: A = transform rows (cams 0-3 / 4-5),
// B = 16 homogeneous points. Each lane ends with full 4-vec results in-lane.
// ------------------------------------------------------------------
__global__ __launch_bounds__(256) void lt_proj_kernel(const float* __restrict__ pc,
                                                      const float* __restrict__ ws,
                                                      const int* __restrict__ bevp,
                                                      float* __restrict__ out) {
    int lane = threadIdx.x & 31;
    int wv   = threadIdx.x >> 5;
    int w    = blockIdx.x * 8 + wv;           // 8 waves per block
    if (w >= BB * GROUPS) return;             // uniform per wave
    int b  = w / GROUPS;
    int g  = w - b * GROUPS;
    int p0 = g * 16;
    int pl = lane & 15;
    bool lo = lane < 16;

    // Load this lane's point (lanes L and L+16 load the same point).
    const float4 pt = *(const float4*)(pc + ((size_t)b * PP + p0 + pl) * 4);

    // B matrix (4x16): row k = component k of [x,y,z,1], col = point.
    v2f Bv;
    Bv.x = lo ? pt.x : pt.z;   // rows 0 / 2
    Bv.y = lo ? pt.y : 1.0f;   // rows 1 / 3

    // A matrix (16x4): row m = (cam m/4, comp m%4); K = weight over [x,y,z,1].
    int cam  = pl >> 2;
    int comp = pl & 3;
    int kb   = lo ? 0 : 2;
    const float* mbase = ws + (size_t)(b * NN) * 16;

    v2f A1;
    {
        const float2 t = *(const float2*)(mbase + cam * 16 + comp * 4 + kb);
        A1.x = t.x; A1.y = t.y;
    }
    v2f A2; A2.x = 0.0f; A2.y = 0.0f;         // rows 8..15 zero (only cams 4,5)
    if (pl < 8) {
        const float2 t = *(const float2*)(mbase + (4 + cam) * 16 + comp * 4 + kb);
        A2.x = t.x; A2.y = t.y;
    }

    v8f cz = {0.f, 0.f, 0.f, 0.f, 0.f, 0.f, 0.f, 0.f};
    // D = A(16x4) x B(4x16) + 0, full FP32
    v8f d1 = __builtin_amdgcn_wmma_f32_16x16x4_f32(false, A1, false, Bv,
                                                   (short)0, cz, false, false);
    v8f d2 = __builtin_amdgcn_wmma_f32_16x16x4_f32(false, A2, false, Bv,
                                                   (short)0, cz, false, false);

    float bh     = 0.5f * (float)(*bevp);                  // bev_side / 2
    float inv_bh = __builtin_amdgcn_rcpf(bh);              // 1/100 here
    float dmax   = bh - 1.0f;
    float iluv   = pt.w;
    int camBase  = lo ? 0 : 2;                // lanes<16: cams{0,1}; lanes>=16: cams{2,3}
    int bn0      = b * NN;

    lt_emit(out, bn0 + camBase + 0, d1[0], d1[1], d1[2], d1[3], iluv, dmax, inv_bh);
    lt_emit(out, bn0 + camBase + 1, d1[4], d1[5], d1[6], d1[7], iluv, dmax, inv_bh);
    if (lo) {   // second WMMA produced cams 4,5 on lanes 0-15 (rows 0..7)
        lt_emit(out, bn0 + 4, d2[0], d2[1], d2[2], d2[3], iluv, dmax, inv_bh);
        lt_emit(out, bn0 + 5, d2[4], d2[5], d2[6], d2[7], iluv, dmax, inv_bh);
    }
}

// ------------------------------------------------------------------
extern "C" void kernel_launch(void* const* d_in, const int* in_sizes, int n_in,
                              void* d_out, int out_size, void* d_ws, size_t ws_size,
                              hipStream_t stream) {
    const float* pcloud = (const float*)d_in[0];   // (B,1,1,P,4) f32
    const float* extr   = (const float*)d_in[1];   // (B,S,N,4,4) f32
    const float* intr   = (const float*)d_in[2];   // (B,S,N,3,3) f32
    const float* view   = (const float*)d_in[3];   // (B,1,1,4,4) f32
    const int*   bev    = (const int*)d_in[4];     // scalar 200
    float* out = (float*)d_out;
    float* ws  = (float*)d_ws;                      // 96*16 floats used

    // 1) zero output grids
    int n4 = out_size >> 2;
    lt_zero_kernel<<<(n4 + 255) / 256, 256, 0, stream>>>(out, out_size);

    // 2) fused per-camera matrices
    lt_mat_kernel<<<1, 128, 0, stream>>>(extr, intr, view, ws);

    // 3) project + scatter (one wave per 16 points per batch)
    int waves = BB * GROUPS;                        // 70000
    lt_proj_kernel<<<(waves + 7) / 8, 256, 0, stream>>>(pcloud, ws, bev, out);
}